// TreeLSTMTagger_71717363909241
// MI455X (gfx1250) — compile-verified
//
#include <hip/hip_runtime.h>
#include <math.h>

// ---------------------------------------------------------------------------
// TreeLSTM tagger for MI455X (gfx1250), wave32, fp32 WMMA (V_WMMA_F32_16X16X4_F32)
// ---------------------------------------------------------------------------

typedef __attribute__((ext_vector_type(2))) float v2f;
typedef __attribute__((ext_vector_type(8))) float v8f;

#define BB   8
#define LLF  128
#define NN   255
#define HH   256
#define NCLS 3
#define HSY  50
#define NINT 127

#define K_L2N 312   // 309 padded to /4
#define K_UP  308   // 306 padded to /4
#define K_DN  612   // already /4
#define LDA   616   // LDS row stride (floats), covers max K

// workspace layout (float offsets)
#define OFF_WL2N  0ull
#define SZ_WL2N   (256ull * 312ull)
#define OFF_WCOMP (OFF_WL2N + SZ_WL2N)
#define SZ_WCOMP  (1024ull * 308ull)
#define OFF_BCOMP (OFF_WCOMP + SZ_WCOMP)
#define OFF_WDEC  (OFF_BCOMP + 1024ull)
#define SZ_WDEC   (2048ull * 612ull)
#define OFF_BDEC  (OFF_WDEC + SZ_WDEC)
#define OFF_STATE (OFF_BDEC + 2048ull)
#define ST_PER_SAMPLE (255ull * 1280ull)   // hU,cU,dh (256 ea) + dc (512)

__device__ __forceinline__ float sigmoidf_(float x) { return 1.0f / (1.0f + __expf(-x)); }

__device__ __forceinline__ v8f wmma4(const float* a, const float* b, v8f acc) {
  v2f av = *(const v2f*)a;
  v2f bv = *(const v2f*)b;
  return __builtin_amdgcn_wmma_f32_16x16x4_f32(false, av, false, bv, (short)0, acc,
                                               false, false);
}

// ---------------------------------------------------------------------------
// Prep: pack padded / concatenated weight matrices + fused biases into ws.
// ---------------------------------------------------------------------------
__global__ void tree_prep_kernel(const float* __restrict__ l2nW,
                                 const float* __restrict__ cWih, const float* __restrict__ cWhh,
                                 const float* __restrict__ cBih, const float* __restrict__ cBhh,
                                 const float* __restrict__ dWih, const float* __restrict__ dWhh,
                                 const float* __restrict__ dBih, const float* __restrict__ dBhh,
                                 float* __restrict__ ws) {
  const size_t total = SZ_WL2N + SZ_WCOMP + 1024ull + SZ_WDEC + 2048ull;
  for (size_t i = (size_t)blockIdx.x * blockDim.x + threadIdx.x; i < total;
       i += (size_t)gridDim.x * blockDim.x) {
    size_t idx = i;
    if (idx < SZ_WL2N) {
      size_t n = idx / 312, k = idx % 312;
      ws[OFF_WL2N + idx] = (k < 309) ? l2nW[n * 309 + k] : 0.0f;
    } else if ((idx -= SZ_WL2N) < SZ_WCOMP) {
      size_t n = idx / 308, k = idx % 308;
      float v = (k < 50) ? cWih[n * 50 + k] : (k < 306 ? cWhh[n * 256 + (k - 50)] : 0.0f);
      ws[OFF_WCOMP + idx] = v;
    } else if ((idx -= SZ_WCOMP) < 1024ull) {
      ws[OFF_BCOMP + idx] = cBih[idx] + cBhh[idx];
    } else if ((idx -= 1024ull) < SZ_WDEC) {
      size_t n = idx / 612, k = idx % 612;
      ws[OFF_WDEC + idx] = (k < 100) ? dWih[n * 100 + k] : dWhh[n * 512 + (k - 100)];
    } else {
      idx -= SZ_WDEC;
      ws[OFF_BDEC + idx] = dBih[idx] + dBhh[idx];
    }
  }
}

// ---------------------------------------------------------------------------
// Main: one workgroup per batch sample; 512 threads = 16 wave32s.
// ---------------------------------------------------------------------------
__global__ __launch_bounds__(512, 1)
void tree_lstm_kernel(const int* __restrict__ x, const int* __restrict__ cue,
                      const int* __restrict__ pos, const int* __restrict__ word_index,
                      const int* __restrict__ left_idx, const int* __restrict__ right_idx,
                      const float* __restrict__ word_emb, const float* __restrict__ syn_emb,
                      const float* __restrict__ l2n_b,
                      const float* __restrict__ grhW, const float* __restrict__ grhB,
                      const float* __restrict__ grcW, const float* __restrict__ grcB,
                      const float* __restrict__ clsW, const float* __restrict__ clsB,
                      float* __restrict__ ws, float* __restrict__ out) {
  __shared__ float sA[16 * LDA];          // staged A tile (16 rows x K)
  __shared__ int sNode[16], sLc[16], sRc[16];

  const int b = blockIdx.x;
  const int tid = threadIdx.x;
  const int lane = tid & 31;
  const int wav = tid >> 5;               // 0..15
  const int r = lane & 15;                // row-within-half / N column
  const int hf = lane >> 4;               // half select (K pair)

  const int* xb = x + b * NN;
  const int* cb = cue + b * NN;
  const int* pb = pos + b * LLF;

  const float* Wl2n = ws + OFF_WL2N;
  const float* Wcmp = ws + OFF_WCOMP;
  const float* Bcmp = ws + OFF_BCOMP;
  const float* Wdec = ws + OFF_WDEC;
  const float* Bdec = ws + OFF_BDEC;
  float* st = ws + OFF_STATE + (size_t)b * ST_PER_SAMPLE;
  float* hU = st;
  float* cU = st + 255 * 256;
  float* dh = st + 2 * 255 * 256;
  float* dc = st + 3 * 255 * 256;

  const float* aBase = sA + r * LDA + 2 * hf;  // per-lane A fragment base

  // ======================= P1: leaf2node (x_emb -> hU leaves) ==============
  for (int rt = 0; rt < 8; ++rt) {
    for (int idx = tid; idx < 16 * K_L2N; idx += 512) {
      int m = idx / K_L2N, k = idx - m * K_L2N;
      int leaf = rt * 16 + m;
      int wi = word_index[leaf];
      float v;
      if (k < 256)      v = word_emb[(size_t)xb[wi] * HH + k];
      else if (k < 259) v = (cb[wi] == (k - 256)) ? 1.0f : 0.0f;
      else if (k < 309) v = syn_emb[pb[leaf] * HSY + (k - 259)];
      else              v = 0.0f;
      sA[m * LDA + k] = v;
    }
    __syncthreads();
    {
      const int col = wav * 16 + r;
      const float* wp = Wl2n + (size_t)col * K_L2N + 2 * hf;
      v8f acc = {};
      for (int k0 = 0; k0 < K_L2N; k0 += 4) {
        __builtin_prefetch(wp + k0 + 64, 0, 0);
        acc = wmma4(aBase + k0, wp + k0, acc);
      }
      const float bias = l2n_b[col];
      for (int j = 0; j < 8; ++j) {
        int node = NINT + rt * 16 + (j + 8 * hf);
        hU[node * HH + col] = acc[j] + bias;
        cU[node * HH + col] = 0.0f;     // c0 = 0 for leaves
      }
    }
    __syncthreads();
  }

  // ======================= P2: upward (level-parallel) =====================
  for (int lev = 6; lev >= 0; --lev) {
    const int nl = 1 << lev, base = nl - 1;
    const int ntiles = (nl + 15) >> 4;
    for (int rt = 0; rt < ntiles; ++rt) {
      if (tid < 16) {
        int k = base + rt * 16 + tid;
        int valid = (k < base + nl);
        sNode[tid] = valid ? k : -1;
        sLc[tid] = valid ? left_idx[k] : 0;
        sRc[tid] = valid ? right_idx[k] : 0;
      }
      __syncthreads();
      for (int idx = tid; idx < 16 * K_UP; idx += 512) {
        int m = idx / K_UP, k = idx - m * K_UP;
        int node = sNode[m];
        float v = 0.0f;
        if (node >= 0) {
          if (k < 50) v = syn_emb[xb[node] * HSY + k];
          else if (k < 306) {
            int j = k - 50;
            v = hU[sLc[m] * HH + j] + hU[sRc[m] * HH + j];
          }
        }
        sA[m * LDA + k] = v;
      }
      __syncthreads();
      {
        const int col = wav * 16 + r;
        const float* wI = Wcmp + (size_t)(col)        * K_UP + 2 * hf;
        const float* wF = Wcmp + (size_t)(col + 256)  * K_UP + 2 * hf;
        const float* wG = Wcmp + (size_t)(col + 512)  * K_UP + 2 * hf;
        const float* wO = Wcmp + (size_t)(col + 768)  * K_UP + 2 * hf;
        v8f aI = {}, aF = {}, aG = {}, aO = {};
        for (int k0 = 0; k0 < K_UP; k0 += 4) {
          v2f av = *(const v2f*)(aBase + k0);
          aI = __builtin_amdgcn_wmma_f32_16x16x4_f32(false, av, false, *(const v2f*)(wI + k0), (short)0, aI, false, false);
          aF = __builtin_amdgcn_wmma_f32_16x16x4_f32(false, av, false, *(const v2f*)(wF + k0), (short)0, aF, false, false);
          aG = __builtin_amdgcn_wmma_f32_16x16x4_f32(false, av, false, *(const v2f*)(wG + k0), (short)0, aG, false, false);
          aO = __builtin_amdgcn_wmma_f32_16x16x4_f32(false, av, false, *(const v2f*)(wO + k0), (short)0, aO, false, false);
        }
        const float bI = Bcmp[col], bF = Bcmp[col + 256];
        const float bG = Bcmp[col + 512], bO = Bcmp[col + 768];
        for (int j = 0; j < 8; ++j) {
          int m = j + 8 * hf;
          int node = sNode[m];
          if (node < 0) continue;
          int l = sLc[m], rr = sRc[m];
          float cp = cU[l * HH + col] + cU[rr * HH + col];
          float gi = sigmoidf_(aI[j] + bI);
          float gf = sigmoidf_(aF[j] + bF);
          float gg = tanhf(aG[j] + bG);
          float go = sigmoidf_(aO[j] + bO);
          float c2 = gf * cp + gi * gg;
          hU[node * HH + col] = go * tanhf(c2);
          cU[node * HH + col] = c2;
        }
      }
      __syncthreads();
    }
  }

  // ======================= P3: global reverse at root ======================
  if (tid < 256) {
    float acc = grhB[tid];
    const float* wr = grhW + tid * 256;
    for (int j = 0; j < 256; ++j) acc += wr[j] * hU[j];
    dh[tid] = tanhf(acc);
  }
  {
    float acc = grcB[tid];
    const float* wr = grcW + tid * 256;
    for (int j = 0; j < 256; ++j) acc += wr[j] * cU[j];
    dc[tid] = tanhf(acc);
  }
  __syncthreads();

  // ======================= P4: downward (level-parallel) ===================
  for (int lev = 0; lev <= 6; ++lev) {
    const int nl = 1 << lev, base = nl - 1;
    const int ntiles = (nl + 15) >> 4;
    for (int rt = 0; rt < ntiles; ++rt) {
      if (tid < 16) {
        int k = base + rt * 16 + tid;
        int valid = (k < base + nl);
        sNode[tid] = valid ? k : -1;
        sLc[tid] = valid ? left_idx[k] : 0;
        sRc[tid] = valid ? right_idx[k] : 0;
      }
      __syncthreads();
      for (int idx = tid; idx < 16 * K_DN; idx += 512) {
        int m = idx / K_DN, k = idx - m * K_DN;
        int node = sNode[m];
        float v = 0.0f;
        if (node >= 0) {
          if (k < 50) {
            int l = sLc[m];
            int si = (l >= NINT) ? pb[l - NINT] : xb[l];
            v = syn_emb[si * HSY + k];
          } else if (k < 100) {
            int rr = sRc[m];
            int si = (rr >= NINT) ? pb[rr - NINT] : xb[rr];
            v = syn_emb[si * HSY + (k - 50)];
          } else if (k < 356) {
            v = dh[node * HH + (k - 100)];
          } else {
            v = hU[node * HH + (k - 356)];
          }
        }
        sA[m * LDA + k] = v;
      }
      __syncthreads();
      for (int t = wav; t < 32; t += 16) {       // 32 col tiles of hidden=512
        const int col = t * 16 + r;
        const float* wI = Wdec + (size_t)(col)         * K_DN + 2 * hf;
        const float* wF = Wdec + (size_t)(col + 512)   * K_DN + 2 * hf;
        const float* wG = Wdec + (size_t)(col + 1024)  * K_DN + 2 * hf;
        const float* wO = Wdec + (size_t)(col + 1536)  * K_DN + 2 * hf;
        v8f aI = {}, aF = {}, aG = {}, aO = {};
        for (int k0 = 0; k0 < K_DN; k0 += 4) {
          v2f av = *(const v2f*)(aBase + k0);
          aI = __builtin_amdgcn_wmma_f32_16x16x4_f32(false, av, false, *(const v2f*)(wI + k0), (short)0, aI, false, false);
          aF = __builtin_amdgcn_wmma_f32_16x16x4_f32(false, av, false, *(const v2f*)(wF + k0), (short)0, aF, false, false);
          aG = __builtin_amdgcn_wmma_f32_16x16x4_f32(false, av, false, *(const v2f*)(wG + k0), (short)0, aG, false, false);
          aO = __builtin_amdgcn_wmma_f32_16x16x4_f32(false, av, false, *(const v2f*)(wO + k0), (short)0, aO, false, false);
        }
        const float bI = Bdec[col], bF = Bdec[col + 512];
        const float bG = Bdec[col + 1024], bO = Bdec[col + 1536];
        for (int j = 0; j < 8; ++j) {
          int m = j + 8 * hf;
          int node = sNode[m];
          if (node < 0) continue;
          int l = sLc[m], rr = sRc[m];
          float cp = dc[node * 512 + col];
          float gi = sigmoidf_(aI[j] + bI);
          float gf = sigmoidf_(aF[j] + bF);
          float gg = tanhf(aG[j] + bG);
          float go = sigmoidf_(aO[j] + bO);
          float c2 = gf * cp + gi * gg;
          float hv = go * tanhf(c2);
          if (col < 256) dh[l * HH + col] = hv;
          else           dh[rr * HH + (col - 256)] = hv;
          dc[l * 512 + col] = c2;
          dc[rr * 512 + col] = c2;
        }
      }
      __syncthreads();
    }
  }

  // ======================= P5: classifier + softmax ========================
  if (tid < NN) {
    float lg[3];
    const float* dhn = dh + tid * HH;
    const float* hun = hU + tid * HH;
    for (int c = 0; c < 3; ++c) {
      float acc = clsB[c];
      const float* wr = clsW + c * 512;
      for (int j = 0; j < 256; ++j) acc += wr[j] * dhn[j];
      for (int j = 0; j < 256; ++j) acc += wr[256 + j] * hun[j];
      lg[c] = tanhf(acc);
    }
    float mx = fmaxf(lg[0], fmaxf(lg[1], lg[2]));
    float e0 = __expf(lg[0] - mx), e1 = __expf(lg[1] - mx), e2 = __expf(lg[2] - mx);
    float s = e0 + e1 + e2;
    float* o = out + ((size_t)b * NN + tid) * NCLS;
    o[0] = e0 / s; o[1] = e1 / s; o[2] = e2 / s;
  }
}

// ---------------------------------------------------------------------------
extern "C" void kernel_launch(void* const* d_in, const int* in_sizes, int n_in,
                              void* d_out, int out_size, void* d_ws, size_t ws_size,
                              hipStream_t stream) {
  (void)in_sizes; (void)n_in; (void)out_size; (void)ws_size;
  const int* x          = (const int*)d_in[0];
  const int* cue        = (const int*)d_in[1];
  const int* pos        = (const int*)d_in[2];
  const int* word_index = (const int*)d_in[3];
  const int* left_idx   = (const int*)d_in[4];
  const int* right_idx  = (const int*)d_in[5];
  const float* word_emb = (const float*)d_in[6];
  const float* syn_emb  = (const float*)d_in[7];
  const float* l2n_W    = (const float*)d_in[8];
  const float* l2n_b    = (const float*)d_in[9];
  const float* comp_Wih = (const float*)d_in[10];
  const float* comp_Whh = (const float*)d_in[11];
  const float* comp_bih = (const float*)d_in[12];
  const float* comp_bhh = (const float*)d_in[13];
  const float* grh_W    = (const float*)d_in[14];
  const float* grh_b    = (const float*)d_in[15];
  const float* grc_W    = (const float*)d_in[16];
  const float* grc_b    = (const float*)d_in[17];
  const float* dec_Wih  = (const float*)d_in[18];
  const float* dec_Whh  = (const float*)d_in[19];
  const float* dec_bih  = (const float*)d_in[20];
  const float* dec_bhh  = (const float*)d_in[21];
  const float* cls_W    = (const float*)d_in[22];
  const float* cls_b    = (const float*)d_in[23];
  float* ws  = (float*)d_ws;
  float* out = (float*)d_out;

  tree_prep_kernel<<<1024, 256, 0, stream>>>(l2n_W, comp_Wih, comp_Whh, comp_bih, comp_bhh,
                                             dec_Wih, dec_Whh, dec_bih, dec_bhh, ws);
  tree_lstm_kernel<<<dim3(BB), dim3(512), 0, stream>>>(
      x, cue, pos, word_index, left_idx, right_idx, word_emb, syn_emb,
      l2n_b, grh_W, grh_b, grc_W, grc_b, cls_W, cls_b, ws, out);
}